// GraphUNetCoarsening_76733885710650
// MI455X (gfx1250) — compile-verified
//
#include <hip/hip_runtime.h>
#include <hip/hip_bf16.h>
#include <math.h>
#include <stdint.h>

// ---- problem constants (match reference) ----
#define BGR    64
#define NNODE  4096
#define SSENT  512
#define CFEAT  128
#define TOTALN (BGR * NNODE)          // 262144
#define NEDGE  4194304
#define K_SENT 3
#define K_OTH  1792                   // ceil(0.5 * (4096-512))
#define KTOT   (K_SENT + K_OTH)       // 1795
#define PTOT   (BGR * KTOT)           // 114880

#define EPB    1024                   // edges per block in scan passes
#define NB     (NEDGE / EPB)          // 4096
static_assert(NEDGE % EPB == 0, "edge blocking");
static_assert(NB == 4096, "scan kernel assumes 4096 block counts");

// d_out float offsets
#define XOUT_OFF   0
#define EROW_OFF   (PTOT * CFEAT)                 // 14,704,640
#define ECOL_OFF   (EROW_OFF + NEDGE)
#define BATCH_OFF  (ECOL_OFF + NEDGE)
#define PERM_OFF   (BATCH_OFF + PTOT)
#define SCOREO_OFF (PERM_OFF + PTOT)
#define NE_OFF     (SCOREO_OFF + PTOT)

typedef __attribute__((ext_vector_type(2))) float v2f;
typedef __attribute__((ext_vector_type(8))) float v8f;

// ---------------------------------------------------------------- norm
__global__ void norm_kernel(const float* __restrict__ w, float* __restrict__ rnorm) {
  __shared__ float red[CFEAT];
  int t = threadIdx.x;
  float v = w[t];
  red[t] = v * v;
  __syncthreads();
  for (int s = CFEAT / 2; s > 0; s >>= 1) {
    if (t < s) red[t] += red[t + s];
    __syncthreads();
  }
  if (t == 0) *rnorm = rsqrtf(red[0]);
}

// ------------------------------------------------- score via WMMA f32 16x16x4
// One wave computes 16 node scores: D = A(16x4 slices of x) * B(w broadcast
// across the 16 columns), accumulated over K=128 in 32 chunks. The 32 weight
// fragments are hoisted into registers (wave-invariant); the K loop is fully
// unrolled and split over two accumulators to break the WMMA->WMMA RAW chain.
// Column N=0 of the 16x16 f32 accumulator (lanes 0/16) holds the dot products.
__global__ void score_wmma_kernel(const float* __restrict__ x,
                                  const float* __restrict__ w,
                                  const float* __restrict__ rnorm_p,
                                  float* __restrict__ score) {
  const int lane   = threadIdx.x & 31;
  const int waveId = (blockIdx.x * blockDim.x + threadIdx.x) >> 5;
  const int rowBase = waveId * 16;
  const int m  = lane & 15;     // matrix row within tile
  const int hi = lane >> 4;     // lanes 16-31 carry K=2,3 of each chunk

  const float* xrow = x + (size_t)(rowBase + m) * CFEAT + 2 * hi;

  // B 4x16 f32 fragments (w broadcast over N): lane<16 -> {w[4k],w[4k+1]},
  // lane>=16 -> {w[4k+2],w[4k+3]}.  Loaded once, kept in VGPRs.
  v2f wreg[CFEAT / 4];
  #pragma unroll
  for (int k = 0; k < CFEAT / 4; ++k)
    wreg[k] = *(const v2f*)(w + 4 * k + 2 * hi);

  v8f acc0 = {0.f, 0.f, 0.f, 0.f, 0.f, 0.f, 0.f, 0.f};
  v8f acc1 = {0.f, 0.f, 0.f, 0.f, 0.f, 0.f, 0.f, 0.f};

  #pragma unroll
  for (int k = 0; k < CFEAT / 4; k += 2) {
    // A 16x4 f32 layout: lane<16 -> {x[m][4k],x[m][4k+1]}, lane>=16 -> {x[m][4k+2],x[m][4k+3]}
    v2f a0 = *(const v2f*)(xrow + 4 * k);
    v2f a1 = *(const v2f*)(xrow + 4 * (k + 1));
    acc0 = __builtin_amdgcn_wmma_f32_16x16x4_f32(false, a0, false, wreg[k],
                                                 (short)0, acc0, false, false);
    acc1 = __builtin_amdgcn_wmma_f32_16x16x4_f32(false, a1, false, wreg[k + 1],
                                                 (short)0, acc1, false, false);
  }

  float rn = *rnorm_p;
  if ((lane & 15) == 0) {                 // lanes 0 (rows 0-7) and 16 (rows 8-15) hold N=0
    int rb = rowBase + hi * 8;
    #pragma unroll
    for (int i = 0; i < 8; ++i) score[rb + i] = tanhf((acc0[i] + acc1[i]) * rn);
  }
}

// ---------------------------------------------------------------- node_map = -1
__global__ void nodemap_init_kernel(int* __restrict__ node_map) {
  int i = blockIdx.x * blockDim.x + threadIdx.x;
  node_map[i] = -1;
}

// ------------------------------------------------- per-graph top-k via ranking
// rank_i = #{ j in segment : s_j > s_i || (s_j == s_i && j < i) }  == position
// in descending stable order, exactly jax.lax.top_k's selection + ordering.
// The 16KB score segment is staged into LDS with gfx1250 async LDS copies.
__global__ void topk_kernel(const float* __restrict__ score, int* __restrict__ perm_int) {
  __shared__ float s[NNODE];
  const int g = blockIdx.x;
  const int t = threadIdx.x;
  const float* sg = score + (size_t)g * NNODE;

  // async global -> LDS: 512 threads x 2 x b128 = 16384 bytes
  {
    unsigned lbase = (unsigned)(uintptr_t)&s[0];
    #pragma unroll
    for (int r = 0; r < 2; ++r) {
      unsigned laddr = lbase + (unsigned)(r * 512 + t) * 16u;
      const float* gaddr = sg + (r * 512 + t) * 4;
      asm volatile("global_load_async_to_lds_b128 %0, %1, off"
                   :: "v"(laddr), "v"(gaddr) : "memory");
    }
    asm volatile("s_wait_asynccnt 0x0" ::: "memory");
  }
  __syncthreads();

  // sentence segment [0, S)
  for (int i = t; i < SSENT; i += blockDim.x) {
    float mi = s[i];
    int r = 0;
    for (int j = 0; j < SSENT; ++j) {
      float sj = s[j];
      r += (sj > mi) || (sj == mi && j < i);
    }
    if (r < K_SENT) perm_int[g * KTOT + r] = g * NNODE + i;
  }
  // other segment [S, N)
  for (int i = SSENT + t; i < NNODE; i += blockDim.x) {
    float mi = s[i];
    int r = 0;
    for (int j = SSENT; j < NNODE; ++j) {
      float sj = s[j];
      r += (sj > mi) || (sj == mi && j < i);
    }
    if (r < K_OTH) perm_int[g * KTOT + K_SENT + r] = g * NNODE + i;
  }
}

// ------------------------------------- scatter node_map + perm/batch/score out
__global__ void scatter_kernel(const int* __restrict__ perm_int,
                               const float* __restrict__ score,
                               int* __restrict__ node_map,
                               float* __restrict__ out_batch,
                               float* __restrict__ out_perm,
                               float* __restrict__ out_score) {
  int i = blockIdx.x * blockDim.x + threadIdx.x;
  if (i >= PTOT) return;
  int p = perm_int[i];
  node_map[p] = i;
  out_perm[i]  = (float)p;
  out_batch[i] = (float)(p >> 12);      // p / NNODE
  out_score[i] = score[p];
}

// ----------------------------------------------------- x_out = x[perm]*score
__global__ void gather_x_kernel(const float4* __restrict__ x4,
                                const int* __restrict__ perm_int,
                                const float* __restrict__ score,
                                float4* __restrict__ xout4) {
  int idx = blockIdx.x * blockDim.x + threadIdx.x;   // over PTOT * (CFEAT/4)
  int row = idx >> 5;                                // CFEAT/4 == 32
  int c4  = idx & 31;
  int p = perm_int[row];
  float s = score[p];
  float4 v = x4[(size_t)p * (CFEAT / 4) + c4];
  v.x *= s; v.y *= s; v.z *= s; v.w *= s;
  xout4[idx] = v;
}

// ---------------------------------------------------- edge scan pass 1: counts
__global__ void edge_pass1_kernel(const int* __restrict__ ei,
                                  const int* __restrict__ node_map,
                                  int* __restrict__ block_counts) {
  __shared__ int red[256];
  const int b = blockIdx.x, t = threadIdx.x;
  const int4* ei4 = (const int4*)ei;
  int4 r4 = ei4[b * 256 + t];                 // rows, 4 consecutive edges
  int4 c4 = ei4[NEDGE / 4 + b * 256 + t];     // cols
  int cnt = 0;
  cnt += (node_map[r4.x] >= 0 && node_map[c4.x] >= 0);
  cnt += (node_map[r4.y] >= 0 && node_map[c4.y] >= 0);
  cnt += (node_map[r4.z] >= 0 && node_map[c4.z] >= 0);
  cnt += (node_map[r4.w] >= 0 && node_map[c4.w] >= 0);
  red[t] = cnt;
  __syncthreads();
  for (int s = 128; s > 0; s >>= 1) {
    if (t < s) red[t] += red[t + s];
    __syncthreads();
  }
  if (t == 0) block_counts[b] = red[0];
}

// -------------------------------------- pass 2: exclusive scan of 4096 counts
__global__ void scan_blocks_kernel(const int* __restrict__ block_counts,
                                   int* __restrict__ block_offsets,
                                   int* __restrict__ total_keep,
                                   float* __restrict__ out_numedges) {
  __shared__ int ts[1024];
  const int t = threadIdx.x;
  int c0 = block_counts[4 * t + 0];
  int c1 = block_counts[4 * t + 1];
  int c2 = block_counts[4 * t + 2];
  int c3 = block_counts[4 * t + 3];
  int sum = c0 + c1 + c2 + c3;
  ts[t] = sum;
  __syncthreads();
  for (int off = 1; off < 1024; off <<= 1) {
    int v = (t >= off) ? ts[t - off] : 0;
    __syncthreads();
    ts[t] += v;
    __syncthreads();
  }
  int excl = ts[t] - sum;
  block_offsets[4 * t + 0] = excl;
  block_offsets[4 * t + 1] = excl + c0;
  block_offsets[4 * t + 2] = excl + c0 + c1;
  block_offsets[4 * t + 3] = excl + c0 + c1 + c2;
  if (t == 1023) {
    *total_keep = ts[1023];
    *out_numedges = (float)ts[1023];
  }
}

// ---------------------------- pass 3: stable scatter (kept front, -1 tail)
__global__ void edge_pass3_kernel(const int* __restrict__ ei,
                                  const int* __restrict__ node_map,
                                  const int* __restrict__ block_offsets,
                                  const int* __restrict__ total_keep,
                                  float* __restrict__ out_row,
                                  float* __restrict__ out_col) {
  __shared__ int pref[256];
  const int b = blockIdx.x, t = threadIdx.x;
  const int4* ei4 = (const int4*)ei;
  int4 r4 = ei4[b * 256 + t];
  int4 c4 = ei4[NEDGE / 4 + b * 256 + t];
  int rm[4], cm[4], kp[4];
  rm[0] = node_map[r4.x]; cm[0] = node_map[c4.x];
  rm[1] = node_map[r4.y]; cm[1] = node_map[c4.y];
  rm[2] = node_map[r4.z]; cm[2] = node_map[c4.z];
  rm[3] = node_map[r4.w]; cm[3] = node_map[c4.w];
  int cnt = 0;
  #pragma unroll
  for (int k = 0; k < 4; ++k) { kp[k] = (rm[k] >= 0 && cm[k] >= 0); cnt += kp[k]; }
  pref[t] = cnt;
  __syncthreads();
  for (int off = 1; off < 256; off <<= 1) {
    int v = (t >= off) ? pref[t - off] : 0;
    __syncthreads();
    pref[t] += v;
    __syncthreads();
  }
  int ke = block_offsets[b] + (pref[t] - cnt);   // global keep-exclusive rank
  const int total = *total_keep;
  const int ebase = b * EPB + t * 4;
  #pragma unroll
  for (int k = 0; k < 4; ++k) {
    int e = ebase + k;
    if (kp[k]) {
      out_row[ke] = (float)rm[k];
      out_col[ke] = (float)cm[k];
      ++ke;
    } else {
      int dpos = total + (e - ke);               // drop-exclusive rank
      out_row[dpos] = -1.0f;
      out_col[dpos] = -1.0f;
    }
  }
}

// ---------------------------------------------------------------- launcher
extern "C" void kernel_launch(void* const* d_in, const int* in_sizes, int n_in,
                              void* d_out, int out_size, void* d_ws, size_t ws_size,
                              hipStream_t stream) {
  const float* x  = (const float*)d_in[0];
  const float* w  = (const float*)d_in[1];
  const int*   ei = (const int*)d_in[2];
  // d_in[3] = batch (recomputed from perm/NNODE)

  float* out = (float*)d_out;

  // workspace layout
  char* ws = (char*)d_ws;
  float* rnorm        = (float*)ws;                                  // 1 f (padded 256B)
  float* score        = (float*)(ws + 256);                          // TOTALN f
  int*   node_map     = (int*)(ws + 256 + 4 * TOTALN);               // TOTALN i
  int*   perm_int     = (int*)(ws + 256 + 8 * TOTALN);               // PTOT i
  int*   block_counts = perm_int + PTOT;                             // NB i
  int*   block_offs   = block_counts + NB;                           // NB i
  int*   total_keep   = block_offs + NB;                             // 1 i

  // 1) rsqrt(||w||^2)
  norm_kernel<<<1, CFEAT, 0, stream>>>(w, rnorm);

  // 2) scores via WMMA: 16 rows/wave, 8 waves/block
  score_wmma_kernel<<<TOTALN / 128, 256, 0, stream>>>(x, w, rnorm, score);

  // 3) node_map = -1
  nodemap_init_kernel<<<TOTALN / 256, 256, 0, stream>>>(node_map);

  // 4) per-graph top-k -> perm_int
  topk_kernel<<<BGR, 512, 0, stream>>>(score, perm_int);

  // 5) scatter new ids + perm/batch/score outputs
  scatter_kernel<<<(PTOT + 255) / 256, 256, 0, stream>>>(
      perm_int, score, node_map, out + BATCH_OFF, out + PERM_OFF, out + SCOREO_OFF);

  // 6) x_out = x[perm] * score[perm]
  gather_x_kernel<<<(PTOT * (CFEAT / 4)) / 256, 256, 0, stream>>>(
      (const float4*)x, perm_int, score, (float4*)(out + XOUT_OFF));

  // 7-9) stable edge compaction
  edge_pass1_kernel<<<NB, 256, 0, stream>>>(ei, node_map, block_counts);
  scan_blocks_kernel<<<1, 1024, 0, stream>>>(block_counts, block_offs, total_keep,
                                             out + NE_OFF);
  edge_pass3_kernel<<<NB, 256, 0, stream>>>(ei, node_map, block_offs, total_keep,
                                            out + EROW_OFF, out + ECOL_OFF);
}